// Voxelization_60636348285361
// MI455X (gfx1250) — compile-verified
//
#include <hip/hip_runtime.h>
#include <climits>
#include <cstdint>

// ---- problem constants (from reference) ----
#define GXN 1408
#define GYN 1600
#define GZN 40
#define LOX 0.0f
#define LOY (-40.0f)
#define LOZ (-3.0f)
#define VSX 0.05f
#define VSY 0.05f
#define VSZ 0.1f
#define MAXV 120000
#define MAXP 10
#define CHUNK 1024          // points per scan block (256 thr x 4 items)
#define MAXBLK 2048         // max scan blocks handled by single-block scan
#define PITER 4             // tiles per block in the async-staged point pass
#define LGRID 256           // blocks for list-based rounds (grid-stride)

// output layout (floats, concatenated in return order)
#define OUT_COORS ((size_t)MAXV * MAXP * 4)   // 4,800,000
#define OUT_NPTS  (OUT_COORS + (size_t)MAXV * 3)
#define OUT_VNUM  (OUT_NPTS + (size_t)MAXV)

// ---- gfx1250 async global->LDS copy (ASYNCcnt-tracked DMA path) ----
#if __has_builtin(__builtin_amdgcn_global_load_async_to_lds_b128)
#define HAS_ASYNC_COPY 1
typedef int vsi4 __attribute__((vector_size(16)));
typedef __attribute__((address_space(1))) vsi4 gvec4;
typedef __attribute__((address_space(3))) vsi4 lvec4;
#endif

__device__ __forceinline__ void async_copy16(const float4* g, float4* l) {
#ifdef HAS_ASYNC_COPY
  __builtin_amdgcn_global_load_async_to_lds_b128((gvec4*)(void*)g, (lvec4*)(void*)l, 0, 0);
#else
  *l = *g;
#endif
}
__device__ __forceinline__ void wait_async_le1() {
#ifdef HAS_ASYNC_COPY
#if __has_builtin(__builtin_amdgcn_s_wait_asynccnt)
  __builtin_amdgcn_s_wait_asynccnt(1);
#else
  asm volatile("s_wait_asynccnt 0x1" ::: "memory");
#endif
#endif
}
__device__ __forceinline__ void wait_async_le0() {
#ifdef HAS_ASYNC_COPY
#if __has_builtin(__builtin_amdgcn_s_wait_asynccnt)
  __builtin_amdgcn_s_wait_asynccnt(0);
#else
  asm volatile("s_wait_asynccnt 0x0" ::: "memory");
#endif
#endif
}

__device__ __forceinline__ unsigned hmix(unsigned h) {
  h ^= h >> 16; h *= 0x85EBCA6Bu;
  h ^= h >> 13; h *= 0xC2B2AE35u;
  h ^= h >> 16; return h;
}

// ---------- init kernels ----------
__global__ void k_zero_f(float* __restrict__ p, int n) {
  int i = blockIdx.x * blockDim.x + threadIdx.x;
  if (i < n) p[i] = 0.0f;
}
__global__ void k_zero_i(int* __restrict__ p, int n) {
  int i = blockIdx.x * blockDim.x + threadIdx.x;
  if (i < n) p[i] = 0;
}
__global__ void k_init_hash(int* __restrict__ key, int* __restrict__ fidx,
                            int* __restrict__ cand, int ht) {
  int i = blockIdx.x * blockDim.x + threadIdx.x;
  if (i < ht) { key[i] = -1; fidx[i] = INT_MAX; cand[i] = INT_MAX; }
}

// ---------- pass 1: async-staged voxelize + hash insert + first-index min ----------
// Each block pipelines PITER tiles of 256 points through a double-buffered LDS
// stage filled by GLOBAL_LOAD_ASYNC_TO_LDS_B128; the ASYNCcnt wait values are
// wave-uniform (issue decisions depend only on wave-base index).
__global__ void k_points(const float4* __restrict__ pts, int n,
                         int* __restrict__ key, int* __restrict__ fidx,
                         int* __restrict__ pent, unsigned ht_mask) {
  __shared__ float4 stage[2][256];
  const int t = threadIdx.x;
  const int tile0 = blockIdx.x * PITER;
  const int wbase = t & ~31;            // wave-uniform lane base within tile

  int pending = 0;
  { // prologue: stage tile 0
    int base = tile0 * 256;
    if (base + wbase < n) {             // wave-uniform issue decision
      int idx = base + t;
      if (idx < n) async_copy16(&pts[idx], &stage[0][t]);
      pending = 1;
    }
  }
  for (int it = 0; it < PITER; ++it) {
    // issue next tile into the other buffer
    if (it + 1 < PITER) {
      int nbase = (tile0 + it + 1) * 256;
      if (nbase + wbase < n) {
        int nidx = nbase + t;
        if (nidx < n) async_copy16(&pts[nidx], &stage[(it + 1) & 1][t]);
        ++pending;
      }
    }
    // wait for the current tile's copy to land
    if (pending >= 2)      wait_async_le1();
    else if (pending == 1) wait_async_le0();
    if (pending > 0) --pending;

    int idx = (tile0 + it) * 256 + t;
    if (idx >= n) continue;
    float4 p = stage[it & 1][t];
    float fx = floorf((p.x - LOX) / VSX);
    float fy = floorf((p.y - LOY) / VSY);
    float fz = floorf((p.z - LOZ) / VSZ);
    int cx = (int)fx, cy = (int)fy, cz = (int)fz;
    bool valid = (fx >= 0.0f) && (fy >= 0.0f) && (fz >= 0.0f) &&
                 (cx < GXN) && (cy < GYN) && (cz < GZN);
    if (!valid) { pent[idx] = -1; continue; }
    int flat = (cz * GYN + cy) * GXN + cx;        // < 2^27
    unsigned s = hmix((unsigned)flat) & ht_mask;
    for (;;) {
      int kk = key[s];
      if (kk == flat) break;
      if (kk == -1) {
        int old = atomicCAS(&key[s], -1, flat);
        if (old == -1 || old == flat) break;
      }
      s = (s + 1u) & ht_mask;
    }
    pent[idx] = (int)s;
    atomicMin(&fidx[s], idx);
  }
}

// ---------- pass 2a: per-block first-point count + ballot mask cache ----------
__global__ void k_scan_blocks(const int* __restrict__ fidx, const int* __restrict__ pent,
                              int n, int* __restrict__ bsum, unsigned* __restrict__ fmask) {
  __shared__ int wsum[32];
  int t = threadIdx.x, b = blockIdx.x;
  int lane = t & 31, wave = t >> 5;
  for (int k = 0; k < 4; ++k) {
    int idx = b * CHUNK + k * 256 + t;
    int f = 0;
    if (idx < n) { int e = pent[idx]; if (e >= 0 && fidx[e] == idx) f = 1; }
    unsigned m = (unsigned)__ballot(f);
    if (lane == 0) {
      wsum[k * 8 + wave] = __popc(m);
      fmask[(size_t)b * 32 + k * 8 + wave] = m;   // cache for pass 2c
    }
  }
  __syncthreads();
  if (t == 0) {
    int r = 0;
    for (int j = 0; j < 32; ++j) r += wsum[j];
    bsum[b] = r;
  }
}

// ---------- pass 2b: single-block exclusive scan of block sums ----------
__global__ void k_scan_offsets(const int* __restrict__ bsum, int* __restrict__ boff,
                               int nblk, int* __restrict__ misc,
                               float* __restrict__ out_vnum) {
  __shared__ int sh[256];
  int t = threadIdx.x;
  int vals[8]; int s = 0;
  for (int k = 0; k < 8; ++k) {
    int j = t * 8 + k;
    int v = (j < nblk) ? bsum[j] : 0;
    vals[k] = v; s += v;
  }
  sh[t] = s;
  __syncthreads();
  for (int off = 1; off < 256; off <<= 1) {
    int v = (t >= off) ? sh[t - off] : 0;
    __syncthreads();
    sh[t] += v;
    __syncthreads();
  }
  int run = sh[t] - s;  // exclusive
  for (int k = 0; k < 8; ++k) {
    int j = t * 8 + k;
    if (j < nblk) boff[j] = run;
    run += vals[k];
  }
  if (t == 255) {
    int total = sh[255];
    int vn = total < MAXV ? total : MAXV;
    misc[0] = vn;
    out_vnum[0] = (float)vn;
  }
}

// ---------- pass 2c (fused): rank + coors + slot-0 claim + loser compaction ----------
// First point of each voxel (flag from cached ballot masks) gets the voxel's
// first-appearance rank, writes coors and voxels[r][0], sets npts[r]=1 (unique
// writer). Valid non-first points are appended to the round-1 worklist.
__global__ void k_rank_claim0(const float4* __restrict__ pts, const int* __restrict__ pent,
                              int n, const int* __restrict__ boff,
                              const unsigned* __restrict__ fmask,
                              int* __restrict__ rank, float* __restrict__ coors,
                              float* __restrict__ voxels, int* __restrict__ npts,
                              int* __restrict__ listA, int* __restrict__ cntA) {
  __shared__ int wsum[32];
  __shared__ int wexc[32];
  int t = threadIdx.x, b = blockIdx.x;
  int lane = t & 31, wave = t >> 5;
  unsigned masks[4];
  for (int k = 0; k < 4; ++k) {
    unsigned m = fmask[(size_t)b * 32 + k * 8 + wave];
    masks[k] = m;
    if (lane == 0) wsum[k * 8 + wave] = __popc(m);
  }
  __syncthreads();
  if (t == 0) {
    int r = 0;
    for (int j = 0; j < 32; ++j) { int v = wsum[j]; wexc[j] = r; r += v; }
  }
  __syncthreads();
  int base = boff[b];
  for (int k = 0; k < 4; ++k) {
    int idx = b * CHUNK + k * 256 + t;
    if (idx >= n) continue;
    unsigned m = masks[k];
    int e = pent[idx];
    if ((m >> lane) & 1u) {
      int r = base + wexc[k * 8 + wave] + __popc(m & ((1u << lane) - 1u));
      rank[e] = r;
      if (r < MAXV) {
        float4 p = pts[idx];
        float fz = floorf((p.z - LOZ) / VSZ);
        float fy = floorf((p.y - LOY) / VSY);
        float fx = floorf((p.x - LOX) / VSX);
        coors[(size_t)r * 3 + 0] = fz;   // (z, y, x) per reference c[:, ::-1]
        coors[(size_t)r * 3 + 1] = fy;
        coors[(size_t)r * 3 + 2] = fx;
        float* dst = voxels + (size_t)r * MAXP * 4;   // slot 0
        dst[0] = p.x; dst[1] = p.y; dst[2] = p.z; dst[3] = p.w;
        npts[r] = 1;                      // unique writer, plain store
      }
    } else if (e >= 0) {
      listA[atomicAdd(cntA, 1)] = idx;    // valid non-first -> round-1 worklist
    }
  }
}

// ---------- rounds 1..9 over compacted worklist (grid-stride on device count) ----------
__global__ void k_cand_list(const int* __restrict__ pent, const int* __restrict__ list,
                            const int* __restrict__ cnt, int* __restrict__ cand) {
  int m = *cnt;
  int stride = gridDim.x * blockDim.x;
  for (int j = blockIdx.x * blockDim.x + threadIdx.x; j < m; j += stride) {
    int i = list[j];
    atomicMin(&cand[pent[i]], i);
  }
}

__global__ void k_claim_list(const float4* __restrict__ pts, const int* __restrict__ pent,
                             const int* __restrict__ list, const int* __restrict__ cnt,
                             int slot, int* __restrict__ cand, const int* __restrict__ rank,
                             float* __restrict__ voxels, int* __restrict__ npts,
                             int* __restrict__ nlist, int* __restrict__ ncnt) {
  int m = *cnt;
  int stride = gridDim.x * blockDim.x;
  for (int j = blockIdx.x * blockDim.x + threadIdx.x; j < m; j += stride) {
    int i = list[j];
    int e = pent[i];
    if (cand[e] == i) {                 // i is this voxel's smallest unclaimed index
      cand[e] = INT_MAX;                // reset for next round (sole writer)
      int r = rank[e];
      if (r < MAXV) {
        float4 p = pts[i];
        float* dst = voxels + ((size_t)r * MAXP + slot) * 4;
        dst[0] = p.x; dst[1] = p.y; dst[2] = p.z; dst[3] = p.w;
        atomicAdd(&npts[r], 1);
      }
    } else {
      nlist[atomicAdd(ncnt, 1)] = i;
    }
  }
}

// ---------- finalize: int counts -> float output ----------
__global__ void k_npts_out(const int* __restrict__ npts, float* __restrict__ out) {
  int v = blockIdx.x * blockDim.x + threadIdx.x;
  if (v < MAXV) out[v] = (float)npts[v];
}

extern "C" void kernel_launch(void* const* d_in, const int* in_sizes, int n_in,
                              void* d_out, int out_size, void* d_ws, size_t ws_size,
                              hipStream_t stream) {
  const int N = in_sizes[0] / 4;                 // points are [N, 4]
  const int nblk = (N + CHUNK - 1) / CHUNK;      // <= MAXBLK for N = 2M
  const float4* pts = (const float4*)d_in[0];
  float* out = (float*)d_out;

  // ---- workspace carve-out; pick hash size by available ws ----
  size_t fixed = (size_t)N * 12 /*pent + listA + listB*/ +
                 (size_t)MAXV * 4 /*npts*/ + (size_t)MAXBLK * 8 /*bsum+boff*/ +
                 (size_t)MAXBLK * 128 /*fmask*/ + 64;
  int htbits = (((size_t)1 << 22) * 16 + fixed <= ws_size) ? 22 : 21;  // 2^21 > worst-case uniques
  size_t HT = (size_t)1 << htbits;
  unsigned ht_mask = (unsigned)(HT - 1);

  char* w = (char*)d_ws;
  int* key   = (int*)w;  w += HT * 4;
  int* fidx  = (int*)w;  w += HT * 4;
  int* rank  = (int*)w;  w += HT * 4;
  int* cand  = (int*)w;  w += HT * 4;
  int* pent  = (int*)w;  w += (size_t)N * 4;
  int* listA = (int*)w;  w += (size_t)N * 4;
  int* listB = (int*)w;  w += (size_t)N * 4;
  int* npts  = (int*)w;  w += (size_t)MAXV * 4;
  int* bsum  = (int*)w;  w += (size_t)MAXBLK * 4;
  int* boff  = (int*)w;  w += (size_t)MAXBLK * 4;
  unsigned* fmask = (unsigned*)w;  w += (size_t)MAXBLK * 128;  // 32 masks per scan block
  int* misc  = (int*)w;                 // [0]=voxel_num, [1]=cntA, [2]=cntB
  int* cntA = misc + 1;
  int* cntB = misc + 2;

  float* voxels = out;
  float* coors  = out + OUT_COORS;
  float* onpts  = out + OUT_NPTS;
  float* ovnum  = out + OUT_VNUM;

  const int TB = 256;
  const int OUT_TOTAL = (int)(OUT_VNUM + 1);
  const int pblk = (N + PITER * 256 - 1) / (PITER * 256);

  // init (every call: harness does not re-poison between replays)
  k_zero_f<<<(OUT_TOTAL + TB - 1) / TB, TB, 0, stream>>>(out, OUT_TOTAL);
  k_init_hash<<<((int)HT + TB - 1) / TB, TB, 0, stream>>>(key, fidx, cand, (int)HT);
  k_zero_i<<<(MAXV + TB - 1) / TB, TB, 0, stream>>>(npts, MAXV);
  k_zero_i<<<1, 32, 0, stream>>>(misc, 3);

  // pass 1: async-staged voxelize + hash insert + first-index min
  k_points<<<pblk, TB, 0, stream>>>(pts, N, key, fidx, pent, ht_mask);

  // pass 2: first-appearance rank scan, fused with slot-0 claim + compaction
  k_scan_blocks<<<nblk, TB, 0, stream>>>(fidx, pent, N, bsum, fmask);
  k_scan_offsets<<<1, TB, 0, stream>>>(bsum, boff, nblk, misc, ovnum);
  k_rank_claim0<<<nblk, TB, 0, stream>>>(pts, pent, N, boff, fmask, rank, coors,
                                         voxels, npts, listA, cntA);

  // rounds 1..9 over shrinking worklists (ping-pong); cand[] is still INT_MAX
  int* lcur = listA; int* ccur = cntA;
  int* lnxt = listB; int* cnxt = cntB;
  for (int s = 1; s < MAXP; ++s) {
    k_cand_list<<<LGRID, TB, 0, stream>>>(pent, lcur, ccur, cand);
    k_zero_i<<<1, 32, 0, stream>>>(cnxt, 1);
    k_claim_list<<<LGRID, TB, 0, stream>>>(pts, pent, lcur, ccur, s, cand, rank,
                                           voxels, npts, lnxt, cnxt);
    int* tl = lcur; lcur = lnxt; lnxt = tl;
    int* tc = ccur; ccur = cnxt; cnxt = tc;
  }

  // finalize counts
  k_npts_out<<<(MAXV + TB - 1) / TB, TB, 0, stream>>>(npts, onpts);
}